// GlobalPointer_38001870635648
// MI455X (gfx1250) — compile-verified
//
#include <hip/hip_runtime.h>
#include <math.h>

// ---- problem constants (from reference) ----
#define NUM_TAGS 9
#define INNER    64
#define HIDDEN   768
#define NOUT     (NUM_TAGS * INNER * 2)   // 1152
#define BATCH    16
#define SEQ      512
#define MASK_VALF 1000000000000.0f

typedef __attribute__((ext_vector_type(2))) float v2f;
typedef __attribute__((ext_vector_type(8))) float v8f;

// -----------------------------------------------------------------------------
// Kernel 1: projection GEMM (enc @ W + b) fused with RoPE, writing
//   qws[b][h][s][d]   (row-major for A-fragment loads in kernel 2)
//   kws[b][h][d][t]   (transposed for coalesced B-fragment loads in kernel 2)
// One wave32 per 16x16 output tile, V_WMMA_F32_16X16X4_F32 over K=768.
// -----------------------------------------------------------------------------
__global__ __launch_bounds__(256) void proj_rope_kernel(
    const float* __restrict__ enc,   // [B*S, HIDDEN]
    const float* __restrict__ Wm,    // [HIDDEN, NOUT]
    const float* __restrict__ bias,  // [NOUT]
    float* __restrict__ qws,         // [B, H, S, INNER]
    float* __restrict__ kws)         // [B, H, INNER, S]
{
    const int wave = blockIdx.x * 8 + (threadIdx.x >> 5);
    const int lane = threadIdx.x & 31;
    const int li   = lane & 15;      // M (A) / N (B,C,D) within tile
    const int kh   = lane >> 4;      // K-half selector for A/B fragments

    const int nTiles = NOUT / 16;    // 72
    const int mTile  = wave / nTiles;
    const int nTile  = wave - mTile * nTiles;
    const int mBase  = mTile * 16;
    const int nBase  = nTile * 16;

    // ---- GEMM: acc(16x16) += A(16x4) * B(4x16), K swept 4 at a time ----
    v8f acc = {};
    const float* __restrict__ arow = enc + (size_t)(mBase + li) * HIDDEN;
    const float* __restrict__ bcol = Wm + (nBase + li);
    #pragma unroll 4
    for (int k0 = 0; k0 < HIDDEN; k0 += 4) {
        const int kk = k0 + 2 * kh;              // even -> 8B-aligned float2
        v2f a = *(const v2f*)(arow + kk);
        v2f bf;
        bf.x = bcol[(size_t)kk * NOUT];
        bf.y = bcol[(size_t)(kk + 1) * NOUT];
        acc = __builtin_amdgcn_wmma_f32_16x16x4_f32(
            /*neg_a=*/false, a, /*neg_b=*/false, bf,
            /*c_mod=*/(short)0, acc, /*reuse_a=*/false, /*reuse_b=*/false);
    }

    // ---- epilogue: bias + RoPE, scatter to q / k^T workspace ----
    const int   n    = nBase + li;               // column in [0, NOUT)
    const float bv   = bias[n];
    const int   tag  = n >> 7;                   // n / 128
    const int   w    = n & 127;
    const int   dd   = w & 63;                   // rotary dim index
    const bool  isK  = (w >= 64);
    // inv_freq = 10000^(-2*(dd/2)/64)
    const float freq = __expf(-(float)(dd & ~1) * (logf(10000.0f) / (float)INNER));
    const float sgn  = (dd & 1) ? 1.0f : -1.0f;  // rotate_half sign

    #pragma unroll
    for (int r = 0; r < 8; ++r) {
        const int m  = mBase + r + (kh << 3);    // C/D row: lanes>=16 carry M=r+8
        const int bb = m >> 9;                   // batch
        const int s  = m & (SEQ - 1);            // sequence position
        float val = acc[r] + bv;
        // rotate_half partner lives in the adjacent lane (n ^ 1)
        float partner = __shfl_xor(val, 1, 32);
        float ang = (float)s * freq;
        float sv = __sinf(ang);
        float cv = __cosf(ang);
        float o  = val * cv + sgn * partner * sv;
        if (isK)
            kws[(((size_t)bb * NUM_TAGS + tag) * INNER + dd) * SEQ + s] = o;
        else
            qws[(((size_t)bb * NUM_TAGS + tag) * SEQ + s) * INNER + dd] = o;
    }
}

// -----------------------------------------------------------------------------
// Kernel 2: per-(b,h) 512x512 QK^T with pad + causal mask + 1/sqrt(d) scale
// fused into the epilogue. One wave32 per 16x16 logits tile, K=64 over WMMA.
// -----------------------------------------------------------------------------
__global__ __launch_bounds__(256) void qk_mask_kernel(
    const float* __restrict__ qws,   // [B, H, S, INNER]
    const float* __restrict__ kws,   // [B, H, INNER, S]
    const int*   __restrict__ amask, // [B, S]
    float* __restrict__ out)         // [B, H, S, S]
{
    const int wave = blockIdx.x * 8 + (threadIdx.x >> 5);
    const int lane = threadIdx.x & 31;
    const int li   = lane & 15;
    const int kh   = lane >> 4;

    const int tilesPerBH = (SEQ / 16) * (SEQ / 16); // 1024
    const int bh   = wave / tilesPerBH;
    const int tile = wave - bh * tilesPerBH;
    const int sT   = tile >> 5;                     // SEQ/16 = 32 tiles per dim
    const int tT   = tile & 31;
    const int sBase = sT * 16;
    const int tBase = tT * 16;
    const int b     = bh / NUM_TAGS;

    const float* __restrict__ q  = qws + (size_t)bh * SEQ * INNER;
    const float* __restrict__ kp = kws + (size_t)bh * INNER * SEQ;

    v8f acc = {};
    const float* __restrict__ qrow = q + (size_t)(sBase + li) * INNER;
    const int t = tBase + li;
    #pragma unroll
    for (int d0 = 0; d0 < INNER; d0 += 4) {
        const int kk = d0 + 2 * kh;                 // even -> aligned float2
        v2f a = *(const v2f*)(qrow + kk);
        v2f bf;
        bf.x = kp[(size_t)kk * SEQ + t];            // lane-coalesced rows
        bf.y = kp[(size_t)(kk + 1) * SEQ + t];
        acc = __builtin_amdgcn_wmma_f32_16x16x4_f32(
            false, a, false, bf, (short)0, acc, false, false);
    }

    const float pad = (float)amask[(size_t)b * SEQ + t];
    const float padPenalty = (1.0f - pad) * MASK_VALF;
    float* __restrict__ orow = out + (size_t)bh * SEQ * SEQ;
    #pragma unroll
    for (int r = 0; r < 8; ++r) {
        const int s = sBase + r + (kh << 3);
        float val = acc[r] * pad - padPenalty;
        if (t < s) val -= MASK_VALF;                // tril(-1) causal mask
        orow[(size_t)s * SEQ + t] = val * 0.125f;   // 1/sqrt(64)
    }
}

// -----------------------------------------------------------------------------
// Host launcher
// inputs: [0] encoder_output f32 (16,512,768), [1] token_ids i64 (unused),
//         [2] attention_mask i32 (16,512), [3] W f32 (768,1152), [4] b f32 (1152)
// output: logits f32 (16,9,512,512)
// -----------------------------------------------------------------------------
extern "C" void kernel_launch(void* const* d_in, const int* in_sizes, int n_in,
                              void* d_out, int out_size, void* d_ws, size_t ws_size,
                              hipStream_t stream) {
    const float* enc   = (const float*)d_in[0];
    const int*   amask = (const int*)  d_in[2];
    const float* Wm    = (const float*)d_in[3];
    const float* bias  = (const float*)d_in[4];
    float*       out   = (float*)d_out;

    float* qws = (float*)d_ws;
    float* kws = qws + (size_t)BATCH * NUM_TAGS * SEQ * INNER; // +18.87 MB

    // Kernel 1: (B*S/16)=512 M-tiles x (NOUT/16)=72 N-tiles = 36864 waves
    const int waves1  = (BATCH * SEQ / 16) * (NOUT / 16);
    proj_rope_kernel<<<waves1 / 8, 256, 0, stream>>>(enc, Wm, bias, qws, kws);

    // Kernel 2: 144 (b,h) x 1024 tiles = 147456 waves
    const int waves2 = BATCH * NUM_TAGS * (SEQ / 16) * (SEQ / 16);
    qk_mask_kernel<<<waves2 / 8, 256, 0, stream>>>(qws, kws, amask, out);
}